// EGNN_19078244729029
// MI455X (gfx1250) — compile-verified
//
#include <hip/hip_runtime.h>
#include <cstdint>
#include <cstddef>

// ---------------------------------------------------------------------------
// EGNN forward for MI455X (gfx1250, wave32).
//   B=4, N=512, D_FEAT=16, D_HIDDEN=64, D_MSG=32, N_LAYERS=4
// Node-level dense layers: V_WMMA_F32_16X16X4_F32, fully unrolled K loops.
// Edge pass: fused online-softmax, hb/pos staged in LDS (bank-conflict-free
// 33-float padded rows), async global->LDS copies (gfx1250 ASYNCcnt path).
// ---------------------------------------------------------------------------

#define B_   4
#define N_   512
#define M_   (B_ * N_)   // 2048 rows
#define H_   64
#define MSG_ 32
#define SEG_ 256         // hb rows staged per LDS segment
#define PAD_ 33          // padded row stride (floats) -> conflict-free reads

typedef float v2f __attribute__((ext_vector_type(2)));
typedef float v8f __attribute__((ext_vector_type(8)));
typedef int   v4i_vs __attribute__((vector_size(16)));   // matches b128 builtin

#if defined(__has_builtin)
#if __has_builtin(__builtin_amdgcn_global_load_async_to_lds_b32) && \
    __has_builtin(__builtin_amdgcn_global_load_async_to_lds_b128)
#define HAVE_ASYNC_LDS 1
#endif
#endif

#ifdef HAVE_ASYNC_LDS
// builtin signatures (from toolchain diagnostics):
//   b32 : (__device__ int*      src, __shared__ int*      dst, Imm off, Imm cpol)
//   b128: (__device__ v4i_vs*   src, __shared__ v4i_vs*   dst, Imm off, Imm cpol)
#define GAS32(p)  ((__attribute__((address_space(1))) int*)(p))
#define LAS32(p)  ((__attribute__((address_space(3))) int*)(p))
#define GAS128(p) ((__attribute__((address_space(1))) v4i_vs*)(p))
#define LAS128(p) ((__attribute__((address_space(3))) v4i_vs*)(p))
__device__ __forceinline__ void wait_async_lds() {
#if defined(__has_builtin) && __has_builtin(__builtin_amdgcn_s_wait_asynccnt)
  __builtin_amdgcn_s_wait_asynccnt(0);
#else
  asm volatile("s_wait_asynccnt 0x0" ::: "memory");
#endif
}
#endif

// ---------------------------------------------------------------------------
// Kernel 1: h = node_feat @ We + be          [2048,16] x [16,64] -> [2048,64]
// ---------------------------------------------------------------------------
__global__ __launch_bounds__(256) void egnn_embed(
    const float* __restrict__ X,     // [M_,16]
    const float* __restrict__ W,     // [16,64]
    const float* __restrict__ bias,  // [64]
    float* __restrict__ Hout)        // [M_,64]
{
  const int lane = threadIdx.x & 31;
  const int wid  = threadIdx.x >> 5;
  const int tile = blockIdx.x * 8 + wid;   // 128 Mtiles * 4 Ntiles = 512
  const int mt = tile >> 2;
  const int nt = tile & 3;
  const int half = lane >> 4;              // 0: K pair (0,1)  1: K pair (2,3)
  const int l    = lane & 15;
  const float* Ar = X + (size_t)(mt * 16 + l) * 16;
  const float* Bc = W + nt * 16 + l;       // ldb = 64
  v8f c = {};
#pragma unroll
  for (int k = 0; k < 16; k += 4) {
    const int kk = k + 2 * half;
    v2f va, vb;
    va.x = Ar[kk];
    va.y = Ar[kk + 1];
    vb.x = Bc[(size_t)kk * 64];
    vb.y = Bc[(size_t)(kk + 1) * 64];
    c = __builtin_amdgcn_wmma_f32_16x16x4_f32(false, va, false, vb,
                                              (short)0, c, false, false);
  }
  const int col = nt * 16 + l;
  const float bv = bias[col];
#pragma unroll
  for (int r = 0; r < 8; ++r) {
    const int row = mt * 16 + r + 8 * half;
    Hout[(size_t)row * H_ + col] = c[r] + bv;
  }
}

// ---------------------------------------------------------------------------
// Kernel 2: hab = h @ [Wa | Wb] + [ba | bb]   [2048,64] x [64,64] -> [2048,64]
//   cols 0..31 = ha, cols 32..63 = hb.  K fully unrolled: 16 WMMAs.
// ---------------------------------------------------------------------------
__global__ __launch_bounds__(256) void egnn_hab(
    const float* __restrict__ h,    // [M_,64]
    const float* __restrict__ Wa,   // [64,32]
    const float* __restrict__ ba,   // [32]
    const float* __restrict__ Wb,   // [64,32]
    const float* __restrict__ bb,   // [32]
    float* __restrict__ hab)        // [M_,64]
{
  const int lane = threadIdx.x & 31;
  const int wid  = threadIdx.x >> 5;
  const int tile = blockIdx.x * 8 + wid;   // 512 tiles
  const int mt = tile >> 2;
  const int nt = tile & 3;
  const int half = lane >> 4;
  const int l    = lane & 15;

  const float* W;
  const float* bi;
  int nl;
  if (nt < 2) { W = Wa; bi = ba; nl = nt * 16; }
  else        { W = Wb; bi = bb; nl = (nt - 2) * 16; }

  const float* Ar = h + (size_t)(mt * 16 + l) * H_;
  const float* Bc = W + nl + l;            // ldb = 32
  v8f c = {};
#pragma unroll
  for (int k = 0; k < H_; k += 4) {
    const int kk = k + 2 * half;
    v2f va, vb;
    va.x = Ar[kk];
    va.y = Ar[kk + 1];
    vb.x = Bc[(size_t)kk * MSG_];
    vb.y = Bc[(size_t)(kk + 1) * MSG_];
    c = __builtin_amdgcn_wmma_f32_16x16x4_f32(false, va, false, vb,
                                              (short)0, c, false, false);
  }
  const int col = nt * 16 + l;
  const float bv = bi[nl + l];
#pragma unroll
  for (int r = 0; r < 8; ++r) {
    const int row = mt * 16 + r + 8 * half;
    hab[(size_t)row * H_ + col] = c[r] + bv;
  }
}

// ---------------------------------------------------------------------------
// Edge-pass LDS staging: copy one 256-row segment of hb (cols 32..63 of hab)
// into padded LDS rows.  Async global->LDS (gfx1250) when available.
// ---------------------------------------------------------------------------
__device__ __forceinline__ void load_hb_seg(const float* __restrict__ habb,
                                            float* __restrict__ shb,
                                            int seg, int tid) {
#ifdef HAVE_ASYNC_LDS
#pragma unroll
  for (int it = 0; it < 32; ++it) {               // 256 rows * 32 cols b32
    const int idx = tid + it * 256;               // 0..8191
    const int row = idx >> 5;
    const int c   = idx & 31;
    const float* src = habb + (size_t)(seg * SEG_ + row) * H_ + MSG_ + c;
    __builtin_amdgcn_global_load_async_to_lds_b32(GAS32(src),
                                                  LAS32(shb + row * PAD_ + c),
                                                  0, 0);
  }
  wait_async_lds();
#else
#pragma unroll
  for (int it = 0; it < 8; ++it) {                // 2048 float4 chunks
    const int idx = tid + it * 256;
    const int row = idx >> 3;
    const int q   = idx & 7;
    const float4 v4 =
        *(const float4*)(habb + (size_t)(seg * SEG_ + row) * H_ + MSG_ + q * 4);
    float* d = shb + row * PAD_ + q * 4;
    d[0] = v4.x; d[1] = v4.y; d[2] = v4.z; d[3] = v4.w;
  }
#endif
}

// ---------------------------------------------------------------------------
// Kernel 3: fused edge pass.  One wave per node i (8 nodes / block),
// lanes = 32 parallel j's.  hb staged in LDS in 2 segments; pos fully
// LDS-resident.  Two passes with online softmax; agg reduced via an LDS
// transpose (reusing the shb region); v via shfl_xor butterflies.
// ---------------------------------------------------------------------------
__global__ __launch_bounds__(256) void egnn_edge(
    const float* __restrict__ hab,   // [B_*N_,64]  (ha | hb)
    const float* __restrict__ pos,   // [B_*N_,3]
    const int*   __restrict__ edges, // [B_,N_,N_]
    const float* __restrict__ vin,   // [B_*N_,3]
    float* __restrict__ vout,        // [B_*N_,3]
    float* __restrict__ agg,         // [B_*N_,32]
    const float* __restrict__ Wd,    // [32]
    const float* __restrict__ bd,    // [32]
    const float* __restrict__ Wat,   // [32]
    const float* __restrict__ bat,   // [1]
    const float* __restrict__ Wx,    // [32]
    const float* __restrict__ bx)    // [1]
{
  __shared__ float            shb[SEG_ * PAD_];   // 33.8 KB, reused as sred
  __shared__ __align__(16) float sposAll[N_ * 3]; // 6 KB
  __shared__ float sWd[32], sbd[32], sWat[32], sWx[32];
  __shared__ float sb2[2];
  __shared__ float sha_s[8][32];

  const int tid  = threadIdx.x;
  const int lane = tid & 31;
  const int w    = tid >> 5;
  const int b    = blockIdx.x >> 6;          // 64 blocks per batch
  const int i    = ((blockIdx.x & 63) << 3) + w;
  const int gi   = b * N_ + i;

  if (w == 0) { sWd[lane]  = Wd[lane];  sbd[lane] = bd[lane]; }
  if (w == 1) { sWat[lane] = Wat[lane]; sWx[lane] = Wx[lane]; }
  if (tid == 64) { sb2[0] = bat[0]; sb2[1] = bx[0]; }
  sha_s[w][lane] = hab[(size_t)gi * H_ + lane];          // ha_i

  const float* posb = pos + (size_t)b * N_ * 3;
  const float* habb = hab + (size_t)b * N_ * H_;

  // stage all of pos[b] into LDS (384 x 16B)
#ifdef HAVE_ASYNC_LDS
#pragma unroll
  for (int it = 0; it < 2; ++it) {
    const int idx = tid + it * 256;
    if (idx < (N_ * 3) / 4) {
      __builtin_amdgcn_global_load_async_to_lds_b128(GAS128(posb + idx * 4),
                                                     LAS128(sposAll + idx * 4),
                                                     0, 0);
    }
  }
  wait_async_lds();
#else
#pragma unroll
  for (int it = 0; it < 2; ++it) {
    const int idx = tid + it * 256;
    if (idx < (N_ * 3) / 4) {
      const float4 v4 = *(const float4*)(posb + idx * 4);
      float* d = sposAll + idx * 4;
      d[0] = v4.x; d[1] = v4.y; d[2] = v4.z; d[3] = v4.w;
    }
  }
#endif
  __syncthreads();

  const float pix = sposAll[i * 3 + 0];
  const float piy = sposAll[i * 3 + 1];
  const float piz = sposAll[i * 3 + 2];
  const float batv = sb2[0], bxv = sb2[1];
  const float* shaw = &sha_s[w][0];
  const int*   erow = edges + (size_t)gi * N_;

  // ---------------- pass 1: online softmax stats ----------------
  float mx = -1e30f, Z = 0.f;
  for (int seg = 0; seg < 2; ++seg) {
    __syncthreads();                     // previous segment's readers done
    load_hb_seg(habb, shb, seg, tid);
    __syncthreads();
    for (int jt = 0; jt < SEG_; jt += 32) {
      const int jl = jt + lane;
      const int j  = seg * SEG_ + jl;
      if (erow[j] > 0) {
        const float dx = pix - sposAll[j * 3 + 0];
        const float dy = piy - sposAll[j * 3 + 1];
        const float dz = piz - sposAll[j * 3 + 2];
        const float d2 = dx * dx + dy * dy + dz * dz;
        const float* hj = shb + jl * PAD_;
        float logit = batv;
#pragma unroll
        for (int c = 0; c < 32; ++c) {
          const float mm = fmaxf(shaw[c] + hj[c] + d2 * sWd[c] + sbd[c], 0.f);
          logit += mm * sWat[c];
        }
        const float nm = fmaxf(mx, logit);
        Z = Z * __expf(mx - nm) + __expf(logit - nm);
        mx = nm;
      }
    }
  }
  // butterfly combine (max, Z) across the 32 lanes
#pragma unroll
  for (int off = 16; off >= 1; off >>= 1) {
    const float om = __shfl_xor(mx, off, 32);
    const float oz = __shfl_xor(Z, off, 32);
    const float nm = fmaxf(mx, om);
    Z = Z * __expf(mx - nm) + oz * __expf(om - nm);
    mx = nm;
  }
  const float inv = (Z > 0.f) ? (1.f / Z) : 0.f;

  // ---------------- pass 2: weighted aggregation ----------------
  float aacc[32];
#pragma unroll
  for (int c = 0; c < 32; ++c) aacc[c] = 0.f;
  float vx = 0.f, vy = 0.f, vz = 0.f;

  for (int seg = 0; seg < 2; ++seg) {
    __syncthreads();
    load_hb_seg(habb, shb, seg, tid);
    __syncthreads();
    for (int jt = 0; jt < SEG_; jt += 32) {
      const int jl = jt + lane;
      const int j  = seg * SEG_ + jl;
      if (erow[j] > 0) {
        const float dx = pix - sposAll[j * 3 + 0];
        const float dy = piy - sposAll[j * 3 + 1];
        const float dz = piz - sposAll[j * 3 + 2];
        const float d2 = dx * dx + dy * dy + dz * dz;
        const float* hj = shb + jl * PAD_;
        float m[32];
        float logit = batv;
#pragma unroll
        for (int c = 0; c < 32; ++c) {
          m[c] = fmaxf(shaw[c] + hj[c] + d2 * sWd[c] + sbd[c], 0.f);
          logit += m[c] * sWat[c];
        }
        const float wgt = __expf(logit - mx) * inv;   // attn weight (e=1)
        float s = bxv;
#pragma unroll
        for (int c = 0; c < 32; ++c) {
          const float wm = m[c] * wgt;
          aacc[c] += wm;
          s += wm * sWx[c];
        }
        const float xw = tanhf(s);
        vx += dx * xw; vy += dy * xw; vz += dz * xw;
      }
    }
  }

  // reduce v across lanes (all lanes end with the total)
#pragma unroll
  for (int off = 16; off >= 1; off >>= 1) {
    vx += __shfl_xor(vx, off, 32);
    vy += __shfl_xor(vy, off, 32);
    vz += __shfl_xor(vz, off, 32);
  }

  // reduce agg (32 ch x 32 lanes) via LDS transpose, reusing shb
  // (8*32*PAD_ == SEG_*PAD_ floats exactly)
  __syncthreads();                         // everyone done reading shb
  float* sred = shb + (size_t)w * 32 * PAD_;
#pragma unroll
  for (int c = 0; c < 32; ++c) sred[lane * PAD_ + c] = aacc[c];
  __syncthreads();
  float asum = 0.f;
#pragma unroll
  for (int jl = 0; jl < 32; ++jl) asum += sred[jl * PAD_ + lane];
  agg[(size_t)gi * MSG_ + lane] = asum;

  if (lane == 0) {
    vout[(size_t)gi * 3 + 0] = vin[(size_t)gi * 3 + 0] + vx;
    vout[(size_t)gi * 3 + 1] = vin[(size_t)gi * 3 + 1] + vy;
    vout[(size_t)gi * 3 + 2] = vin[(size_t)gi * 3 + 2] + vz;
  }
}

// ---------------------------------------------------------------------------
// Kernel 4: h' = h + relu(concat(h, agg) @ Wn + bn)    K = 96 -> 24 WMMAs
// ---------------------------------------------------------------------------
__global__ __launch_bounds__(256) void egnn_node(
    const float* __restrict__ h,    // [M_,64]
    const float* __restrict__ agg,  // [M_,32]
    const float* __restrict__ Wn,   // [96,64]
    const float* __restrict__ bn,   // [64]
    float* __restrict__ hout)       // [M_,64]
{
  const int lane = threadIdx.x & 31;
  const int wid  = threadIdx.x >> 5;
  const int tile = blockIdx.x * 8 + wid;   // 512 tiles
  const int mt = tile >> 2;
  const int nt = tile & 3;
  const int half = lane >> 4;
  const int l    = lane & 15;
  const int arow = mt * 16 + l;
  v8f c = {};
#pragma unroll
  for (int k = 0; k < 96; k += 4) {
    const int kk = k + 2 * half;       // even; kk,kk+1 stay on one side of 64
    v2f va, vb;
    if (kk < 64) {
      const float* p = h + (size_t)arow * H_ + kk;
      va.x = p[0]; va.y = p[1];
    } else {
      const float* p = agg + (size_t)arow * MSG_ + (kk - 64);
      va.x = p[0]; va.y = p[1];
    }
    vb.x = Wn[(size_t)kk * H_ + nt * 16 + l];
    vb.y = Wn[(size_t)(kk + 1) * H_ + nt * 16 + l];
    c = __builtin_amdgcn_wmma_f32_16x16x4_f32(false, va, false, vb,
                                              (short)0, c, false, false);
  }
  const int col = nt * 16 + l;
  const float bv = bn[col];
#pragma unroll
  for (int r = 0; r < 8; ++r) {
    const int row = mt * 16 + r + 8 * half;
    const float t = fmaxf(c[r] + bv, 0.f);
    hout[(size_t)row * H_ + col] = h[(size_t)row * H_ + col] + t;
  }
}

// ---------------------------------------------------------------------------
// Kernel 5: out = pos + v
// ---------------------------------------------------------------------------
__global__ __launch_bounds__(256) void egnn_final(
    const float* __restrict__ pos, const float* __restrict__ v,
    float* __restrict__ out, int n)
{
  const int idx = blockIdx.x * blockDim.x + threadIdx.x;
  if (idx < n) out[idx] = pos[idx] + v[idx];
}

// ---------------------------------------------------------------------------
// Launch.  Param pytree flattens with dict keys sorted (jax convention):
//   d_in: 0 node_feat, 1 pos, 2 vel, 3 edges, 4 embed.W, 5 embed.b,
//   then per layer l (stride 12, base 6):
//     +0 Wa  +1 Wattn  +2 Wb  +3 Wd  +4 Wn  +5 Wx
//     +6 ba  +7 battn  +8 bb  +9 bd  +10 bn +11 bx
// ---------------------------------------------------------------------------
extern "C" void kernel_launch(void* const* d_in, const int* in_sizes, int n_in,
                              void* d_out, int out_size, void* d_ws, size_t ws_size,
                              hipStream_t stream) {
  (void)in_sizes; (void)n_in; (void)out_size; (void)ws_size;

  const float* node_feat = (const float*)d_in[0];
  const float* pos       = (const float*)d_in[1];
  const float* vel       = (const float*)d_in[2];
  const int*   edges     = (const int*)d_in[3];
  const float* We        = (const float*)d_in[4];
  const float* be        = (const float*)d_in[5];

  // workspace layout (floats), ~1.86 MB total
  float* ws   = (float*)d_ws;
  float* hA   = ws;                       // [2048,64]
  float* hB   = hA  + (size_t)M_ * H_;    // [2048,64]
  float* hab  = hB  + (size_t)M_ * H_;    // [2048,64]
  float* aggb = hab + (size_t)M_ * H_;    // [2048,32]
  float* vbuf = aggb + (size_t)M_ * MSG_; // [2048,3]

  egnn_embed<<<64, 256, 0, stream>>>(node_feat, We, be, hA);

  const float* vin = vel;
  float* hcur = hA;
  float* hnext = hB;

  for (int l = 0; l < 4; ++l) {
    const int p = 6 + 12 * l;
    const float* Wa  = (const float*)d_in[p + 0];
    const float* Wat = (const float*)d_in[p + 1];
    const float* Wb  = (const float*)d_in[p + 2];
    const float* Wd  = (const float*)d_in[p + 3];
    const float* Wn  = (const float*)d_in[p + 4];
    const float* Wx  = (const float*)d_in[p + 5];
    const float* ba  = (const float*)d_in[p + 6];
    const float* bat = (const float*)d_in[p + 7];
    const float* bb  = (const float*)d_in[p + 8];
    const float* bd  = (const float*)d_in[p + 9];
    const float* bn  = (const float*)d_in[p + 10];
    const float* bx  = (const float*)d_in[p + 11];

    egnn_hab<<<64, 256, 0, stream>>>(hcur, Wa, ba, Wb, bb, hab);
    egnn_edge<<<B_ * (N_ / 8), 256, 0, stream>>>(hab, pos, edges, vin, vbuf,
                                                 aggb, Wd, bd, Wat, bat, Wx, bx);
    egnn_node<<<64, 256, 0, stream>>>(hcur, aggb, Wn, bn, hnext);

    vin = vbuf;
    float* tmp = hcur; hcur = hnext; hnext = tmp;
  }

  egnn_final<<<(M_ * 3 + 255) / 256, 256, 0, stream>>>(pos, vbuf, (float*)d_out,
                                                       M_ * 3);
}